// CrfDecoderABC_2559800508544
// MI455X (gfx1250) — compile-verified
//
#include <hip/hip_runtime.h>
#include <math.h>

// CRF forward (log-partition), B=64, L=512, C=1, N=128.
// Exp-space reformulation: each sequential step becomes a real GEMM on
// V_WMMA_F32_16X16X4_F32:
//   m = rowmax(alpha); p = exp(alpha-m); s = p @ exp(T); alpha' = log(s)+m+e_t
// 4 workgroups x 16 batches (== WMMA M), 8 wave32s each, one 16x16 output
// tile per wave. 3 workgroup barriers per time step.

typedef __attribute__((ext_vector_type(2))) float v2f;
typedef __attribute__((ext_vector_type(8))) float v8f;

constexpr int Bb   = 64;
constexpr int Ll   = 512;
constexpr int Nn   = 128;
constexpr int ROWS = 16;   // batches per workgroup (== WMMA M)

__launch_bounds__(256, 1)
__global__ void crf_forward_kernel(const float* __restrict__ emissions,     // [B,L,1,N]
                                   const float* __restrict__ transitions,   // [1,N,N]
                                   const float* __restrict__ start_t,       // [1,N]
                                   const float* __restrict__ end_t,         // [1,N]
                                   const int*   __restrict__ lengths,       // [B]
                                   float*       __restrict__ out)           // [B,1]
{
    __shared__ float expT[Nn][Nn];      // 64 KB: exp(transitions), loaded once
    __shared__ float alpha[ROWS][Nn];   // 8 KB : running log-alphas
    __shared__ float pmat[ROWS][Nn];    // 8 KB : exp(alpha - rowmax)
    __shared__ float pmax[ROWS][16];    // partial row maxima / sums
    __shared__ float rmax[ROWS];        // row maxima (for GEMM epilogue)
    __shared__ int   slen[ROWS];

    const int tid = threadIdx.x;
    const int bb  = blockIdx.x * ROWS;  // batch base for this workgroup

    // ---- one-time setup ------------------------------------------------
    for (int idx = tid; idx < Nn * Nn; idx += 256)
        (&expT[0][0])[idx] = __expf(transitions[idx]);

    if (tid < ROWS) slen[tid] = lengths[bb + tid];

    for (int idx = tid; idx < ROWS * Nn; idx += 256) {
        int r = idx >> 7, j = idx & (Nn - 1);
        alpha[r][j] = start_t[j] + emissions[(size_t)(bb + r) * Ll * Nn + j];
    }
    __syncthreads();

    const int lane   = tid & 31;
    const int wave   = tid >> 5;        // 0..7 -> output column tile
    const int lane16 = lane & 15;
    const int hi     = lane >> 4;       // half-wave selector (K / M split)
    const int colN   = wave * 16 + lane16;

    const int rrow = tid >> 4;          // reduction mapping: 16 threads/row
    const int rcol = tid & 15;

    // ---- sequential time loop ------------------------------------------
    for (int t = 1; t < Ll; ++t) {
        // Preload this step's emissions for our C-tile slots; latency hides
        // under the max/exp phases below.
        float ebuf[8];
        #pragma unroll
        for (int r = 0; r < 8; ++r) {
            int M = r + hi * 8;
            ebuf[r] = emissions[((size_t)(bb + M) * Ll + t) * Nn + colN];
        }

        // Phase A: partial row max; keep alpha values in registers.
        float av[8];
        float mx = -INFINITY;
        #pragma unroll
        for (int k = 0; k < 8; ++k) {
            av[k] = alpha[rrow][rcol + 16 * k];
            mx = fmaxf(mx, av[k]);
        }
        pmax[rrow][rcol] = mx;
        __syncthreads();

        // Phase B: every thread reduces its row's 16 partials itself (no
        // extra barrier); publish rmax for the epilogue; exp from registers.
        float rowmx = -INFINITY;
        #pragma unroll
        for (int k = 0; k < 16; ++k) rowmx = fmaxf(rowmx, pmax[rrow][k]);
        if (rcol == 0) rmax[rrow] = rowmx;
        #pragma unroll
        for (int k = 0; k < 8; ++k)
            pmat[rrow][rcol + 16 * k] = __expf(av[k] - rowmx);
        __syncthreads();

        // Phase C: GEMM [16x128] @ [128x128]; this wave owns columns
        // [16*wave, 16*wave+16). K marched in 32 steps of 4.
        // A 16x4 f32: v0 = K0 (lanes 0-15) / K2 (16-31); v1 = K1/K3.
        // B 4x16 f32 mirrored; C 16x16 f32: lanes 16-31 hold M=8..15.
        v8f acc = {};
        #pragma unroll
        for (int kb = 0; kb < 32; ++kb) {
            const int k0 = kb * 4 + hi * 2;
            v2f a, b;
            a.x = pmat[lane16][k0];
            a.y = pmat[lane16][k0 + 1];
            b.x = expT[k0][colN];
            b.y = expT[k0 + 1][colN];
            acc = __builtin_amdgcn_wmma_f32_16x16x4_f32(
                false, a, false, b, (short)0, acc, false, false);
        }

        // Epilogue: alpha' = log(s) + m + e_t, frozen past each length.
        #pragma unroll
        for (int r = 0; r < 8; ++r) {
            const int M    = r + hi * 8;
            const float oldv = alpha[M][colN];
            const float nv   = __logf(acc[r]) + rmax[M] + ebuf[r];
            alpha[M][colN]   = (t < slen[M]) ? nv : oldv;
        }
        __syncthreads();
    }

    // ---- termination: out[b] = logsumexp_j(alpha[b][j] + end[j]) --------
    float tv[8];
    float mx = -INFINITY;
    #pragma unroll
    for (int k = 0; k < 8; ++k) {
        tv[k] = alpha[rrow][rcol + 16 * k] + end_t[rcol + 16 * k];
        mx = fmaxf(mx, tv[k]);
    }
    pmax[rrow][rcol] = mx;
    __syncthreads();
    float rowmx = -INFINITY;
    #pragma unroll
    for (int k = 0; k < 16; ++k) rowmx = fmaxf(rowmx, pmax[rrow][k]);
    __syncthreads();              // pmax reused below for sums
    float s = 0.f;
    #pragma unroll
    for (int k = 0; k < 8; ++k) s += __expf(tv[k] - rowmx);
    pmax[rrow][rcol] = s;
    if (rcol == 0) rmax[rrow] = rowmx;
    __syncthreads();
    if (tid < ROWS) {
        float acc = 0.f;
        #pragma unroll
        for (int k = 0; k < 16; ++k) acc += pmax[tid][k];
        out[bb + tid] = __logf(acc) + rmax[tid];
    }
}

extern "C" void kernel_launch(void* const* d_in, const int* in_sizes, int n_in,
                              void* d_out, int out_size, void* d_ws, size_t ws_size,
                              hipStream_t stream) {
    (void)in_sizes; (void)n_in; (void)out_size; (void)d_ws; (void)ws_size;
    const float* emissions   = (const float*)d_in[0];
    const float* transitions = (const float*)d_in[1];
    const float* start_t     = (const float*)d_in[2];
    const float* end_t       = (const float*)d_in[3];
    const int*   lengths     = (const int*)d_in[4];
    float* out = (float*)d_out;

    dim3 grid(Bb / ROWS);   // 4 workgroups, 16 batch rows each
    dim3 block(256);        // 8 wave32s
    crf_forward_kernel<<<grid, block, 0, stream>>>(
        emissions, transitions, start_t, end_t, lengths, out);
}